// LightweightDCNv2_78520592105639
// MI455X (gfx1250) — compile-verified
//
#include <hip/hip_runtime.h>
#include <hip/hip_bf16.h>
#include <math.h>

// Problem constants (from reference): B=16, C=256, H=W=64, Cout=256
#define BN    16
#define CIN   256
#define HH    64
#define WW    64
#define HW    4096          // H*W
#define COUT  256
#define NPOS  65536         // B*H*W

typedef float v2f __attribute__((ext_vector_type(2)));
typedef float v8f __attribute__((ext_vector_type(8)));

// ---------------------------------------------------------------------------
// K0: NCHW -> NHWC transpose of x.  Per image: (C=256) x (HW=4096) matrix
// transpose via 32x32 LDS tiles. 256 threads/block.
// ---------------------------------------------------------------------------
__global__ __launch_bounds__(256) void k_transpose(const float* __restrict__ x,
                                                   float* __restrict__ xT) {
  __shared__ float tile[32][33];
  int bid = blockIdx.x;
  int hwT = bid & 127;          // 4096/32 tiles along hw
  int cT  = (bid >> 7) & 7;     // 256/32 tiles along c
  int b   = bid >> 10;
  int tx  = threadIdx.x & 31;
  int ty  = threadIdx.x >> 5;   // 8 rows of 32
  const float* src = x + ((size_t)b * CIN + (size_t)cT * 32) * HW + (size_t)hwT * 32;
#pragma unroll
  for (int i = 0; i < 32; i += 8)
    tile[ty + i][tx] = src[(size_t)(ty + i) * HW + tx];
  __syncthreads();
  float* dst = xT + ((size_t)b * HW + (size_t)hwT * 32) * CIN + (size_t)cT * 32;
#pragma unroll
  for (int i = 0; i < 32; i += 8)
    dst[(size_t)(ty + i) * CIN + tx] = tile[tx][ty + i];
}

// ---------------------------------------------------------------------------
// K1: offset conv (3x3, C=256 -> 3) + sigmoid + coordinate generation.
// One wave (32 lanes) per output pixel; 8 waves / block.
// Weights staged in LDS as [j][tap][c] for coalesced lane access.
// ---------------------------------------------------------------------------
__global__ __launch_bounds__(256) void k_offset(const float* __restrict__ xT,
                                                const float* __restrict__ w_off,
                                                const float* __restrict__ b_off,
                                                float* __restrict__ ixp,
                                                float* __restrict__ iyp,
                                                float* __restrict__ mkp) {
  __shared__ float wsm[3 * 9 * 256];  // 27 KB
  // w_off flat layout: j*2304 + c*9 + t   (t = dy*3+dx)
  for (int i = threadIdx.x; i < 3 * 256 * 9; i += 256) {
    int j = i / 2304;
    int rem = i - j * 2304;
    int c = rem / 9;
    int t = rem - c * 9;
    wsm[(j * 9 + t) * 256 + c] = w_off[i];
  }
  __syncthreads();

  int wave = threadIdx.x >> 5;
  int lane = threadIdx.x & 31;
  int pos = blockIdx.x * 8 + wave;      // 0..65535
  int b = pos >> 12;
  int hw = pos & 4095;
  int h = hw >> 6;
  int w = hw & 63;

  float a0 = 0.f, a1 = 0.f, a2 = 0.f;
#pragma unroll
  for (int t = 0; t < 9; t++) {
    int y  = h + t / 3 - 1;
    int xx = w + t % 3 - 1;
    if (y < 0 || y >= HH || xx < 0 || xx >= WW) continue;
    const float* src = xT + (((size_t)b * HH + y) * WW + xx) * CIN;
    const float* w0 = &wsm[(0 * 9 + t) * 256];
    const float* w1 = &wsm[(1 * 9 + t) * 256];
    const float* w2 = &wsm[(2 * 9 + t) * 256];
#pragma unroll
    for (int q = 0; q < 8; q++) {
      int cc = lane + q * 32;
      float v = src[cc];
      a0 = fmaf(v, w0[cc], a0);
      a1 = fmaf(v, w1[cc], a1);
      a2 = fmaf(v, w2[cc], a2);
    }
  }
  // wave32 reduction
#pragma unroll
  for (int off = 16; off > 0; off >>= 1) {
    a0 += __shfl_down(a0, off, 32);
    a1 += __shfl_down(a1, off, 32);
    a2 += __shfl_down(a2, off, 32);
  }
  if (lane == 0) {
    float offy = a0 + b_off[0];
    float offx = a1 + b_off[1];
    float m = 1.0f / (1.0f + __expf(-(a2 + b_off[2])));
    // gx = -1 + 2*w/(W-1) + offx*2/W ;  ix = (gx+1)*0.5*(W-1)
    float gx = -1.0f + 2.0f * (float)w / 63.0f + offx * (2.0f / 64.0f);
    float gy = -1.0f + 2.0f * (float)h / 63.0f + offy * (2.0f / 64.0f);
    ixp[pos] = (gx + 1.0f) * 0.5f * 63.0f;
    iyp[pos] = (gy + 1.0f) * 0.5f * 63.0f;
    mkp[pos] = m;
  }
}

// ---------------------------------------------------------------------------
// K2: bilinear grid-sample (zeros padding, align_corners) * mask -> NHWC x_def
// Block = one pixel, thread = channel (coalesced gathers per corner).
// ---------------------------------------------------------------------------
__global__ __launch_bounds__(256) void k_sample(const float* __restrict__ xT,
                                                const float* __restrict__ ixp,
                                                const float* __restrict__ iyp,
                                                const float* __restrict__ mkp,
                                                float* __restrict__ xdef) {
  int pos = blockIdx.x;
  int c = threadIdx.x;
  int b = pos >> 12;
  float fx = ixp[pos], fy = iyp[pos], m = mkp[pos];
  float x0f = floorf(fx), y0f = floorf(fy);
  float wx1 = fx - x0f, wx0 = 1.0f - wx1;
  float wy1 = fy - y0f, wy0 = 1.0f - wy1;
  int x0 = (int)x0f, y0 = (int)y0f;
  const float* base = xT + ((size_t)b << 12) * CIN;

  auto corner = [&](int yy, int xx, float wgt) -> float {
    bool valid = (xx >= 0) && (xx <= WW - 1) && (yy >= 0) && (yy <= HH - 1);
    int xi = min(max(xx, 0), WW - 1);
    int yi = min(max(yy, 0), HH - 1);
    float v = base[((size_t)(yi << 6) + xi) * CIN + c];
    return valid ? v * wgt : 0.0f;
  };

  float acc = corner(y0, x0, wy0 * wx0) + corner(y0, x0 + 1, wy0 * wx1) +
              corner(y0 + 1, x0, wy1 * wx0) + corner(y0 + 1, x0 + 1, wy1 * wx1);
  xdef[(size_t)pos * CIN + c] = acc * m;
}

// ---------------------------------------------------------------------------
// K3: depthwise 3x3 conv on NHWC x_def -> NHWC dw output.
// Thread = channel; block handles 16 consecutive pixels of one row.
// ---------------------------------------------------------------------------
__global__ __launch_bounds__(256) void k_dw(const float* __restrict__ xdef,
                                            const float* __restrict__ w_dw,
                                            float* __restrict__ dwo) {
  int c = threadIdx.x;
  float wr[9];
#pragma unroll
  for (int t = 0; t < 9; t++) wr[t] = w_dw[c * 9 + t];
  int pos0 = blockIdx.x * 16;
  int b = pos0 >> 12;
  for (int p = 0; p < 16; p++) {
    int pos = pos0 + p;
    int hw = pos & 4095;
    int h = hw >> 6;
    int w = hw & 63;
    float acc = 0.0f;
#pragma unroll
    for (int t = 0; t < 9; t++) {
      int y  = h + t / 3 - 1;
      int xx = w + t % 3 - 1;
      if (y >= 0 && y < HH && xx >= 0 && xx < WW)
        acc = fmaf(xdef[(((size_t)b << 12) + (y << 6) + xx) * CIN + c], wr[t], acc);
    }
    dwo[(size_t)pos * CIN + c] = acc;
  }
}

// ---------------------------------------------------------------------------
// K4: pointwise 1x1 conv as WMMA f32 GEMM.
//   D[m=cout][n=spatial] = sum_k w_pw[m][k] * dw_nhwc[n][k]
// Each wave: 32x32 output tile = 4 accumulators of V_WMMA_F32_16X16X4_F32.
// A layout (16x4 f32): lane<16 -> M=lane, K=k+{0,1}; lane>=16 -> K=k+{2,3}
// B layout (4x16 f32): lane<16 -> N=lane, K=k+{0,1}(v0,v1); lane>=16 -> K=k+{2,3}
// Both operands are a contiguous float2 at row*256 + k + 2*(lane>>4).
// C/D layout: VGPR r -> M = r + 8*(lane>>4), N = lane&15.
// ---------------------------------------------------------------------------
__global__ __launch_bounds__(256) void k_pw_gemm(const float* __restrict__ Bm,
                                                 const float* __restrict__ w_pw,
                                                 float* __restrict__ out) {
  int wave = ((blockIdx.x * 256) + threadIdx.x) >> 5;
  int lane = threadIdx.x & 31;
  int tile_m = wave & 7;        // 256/32 cout tiles
  int tile_n = wave >> 3;       // 65536/32 spatial tiles
  int cout0 = tile_m << 5;
  int s0 = tile_n << 5;
  int half = lane >> 4;
  int nl = lane & 15;

  const float* A0 = w_pw + (size_t)(cout0 + nl) * CIN + (half << 1);
  const float* A1 = A0 + 16 * CIN;
  const float* B0 = Bm + (size_t)(s0 + nl) * CIN + (half << 1);
  const float* B1 = B0 + 16 * CIN;

  v8f acc00 = {}, acc01 = {}, acc10 = {}, acc11 = {};

#pragma unroll 8
  for (int k = 0; k < CIN; k += 4) {
    v2f a0 = *(const v2f*)(A0 + k);
    v2f a1 = *(const v2f*)(A1 + k);
    v2f b0 = *(const v2f*)(B0 + k);
    v2f b1 = *(const v2f*)(B1 + k);
    acc00 = __builtin_amdgcn_wmma_f32_16x16x4_f32(false, a0, false, b0, (short)0, acc00, false, false);
    acc01 = __builtin_amdgcn_wmma_f32_16x16x4_f32(false, a0, false, b1, (short)0, acc01, false, false);
    acc10 = __builtin_amdgcn_wmma_f32_16x16x4_f32(false, a1, false, b0, (short)0, acc10, false, false);
    acc11 = __builtin_amdgcn_wmma_f32_16x16x4_f32(false, a1, false, b1, (short)0, acc11, false, false);
  }

  int b = s0 >> 12;
  int hw0 = s0 & 4095;
  float* ob = out + ((size_t)b << 20);  // b * Cout * HW
#pragma unroll
  for (int r = 0; r < 8; r++) {
    int mrow = r + (half << 3);
    ob[(size_t)(cout0 + mrow) * HW + hw0 + nl]           = acc00[r];
    ob[(size_t)(cout0 + mrow) * HW + hw0 + 16 + nl]      = acc01[r];
    ob[(size_t)(cout0 + 16 + mrow) * HW + hw0 + nl]      = acc10[r];
    ob[(size_t)(cout0 + 16 + mrow) * HW + hw0 + 16 + nl] = acc11[r];
  }
}

// ---------------------------------------------------------------------------
extern "C" void kernel_launch(void* const* d_in, const int* in_sizes, int n_in,
                              void* d_out, int out_size, void* d_ws, size_t ws_size,
                              hipStream_t stream) {
  const float* x     = (const float*)d_in[0];
  const float* w_off = (const float*)d_in[1];
  const float* b_off = (const float*)d_in[2];
  const float* w_dw  = (const float*)d_in[3];
  const float* w_pw  = (const float*)d_in[4];
  float* out = (float*)d_out;

  // Workspace layout (floats):
  //   xT   : 16,777,216   (NHWC input; later reused as dw output)
  //   xdef : 16,777,216   (NHWC deformed+masked)
  //   ix/iy/mask : 3 * 65,536
  float* xT    = (float*)d_ws;
  float* xdef  = xT + (size_t)16777216;
  float* ixp   = xdef + (size_t)16777216;
  float* iyp   = ixp + 65536;
  float* mkp   = iyp + 65536;
  float* dwout = xT;  // alias: xT is dead after k_sample

  k_transpose<<<16384, 256, 0, stream>>>(x, xT);
  k_offset<<<NPOS / 8, 256, 0, stream>>>(xT, w_off, b_off, ixp, iyp, mkp);
  k_sample<<<NPOS, 256, 0, stream>>>(xT, ixp, iyp, mkp, xdef);
  k_dw<<<NPOS / 16, 256, 0, stream>>>(xdef, w_dw, dwout);
  k_pw_gemm<<<(8 * (NPOS / 32)) / 8, 256, 0, stream>>>(dwout, w_pw, out);
}